// GCN_72851235274901
// MI455X (gfx1250) — compile-verified
//
#include <hip/hip_runtime.h>

// GCN 2-layer forward for MI455X (gfx1250, wave32).
//   pre1 = X @ W1            (f32 WMMA 16x16x4)
//   h    = spmm(A, pre1)     (edge-parallel, global_atomic_add_f32)
//   pre2 = relu(h) @ W2      (f32 WMMA, ReLU fused into A-load)
//   out  = spmm(A, pre2)
//
// N=50000 (multiple of 16), F=512, H=128, C=32, E=800000.

#define NFEAT 512
#define NHID  128
#define NCLS  32

typedef __attribute__((ext_vector_type(2))) float v2f;
typedef __attribute__((ext_vector_type(8))) float v8f;

// ---------------------------------------------------------------------------
// zero-fill (float4 granular)
// ---------------------------------------------------------------------------
__global__ __launch_bounds__(256) void gcn_zero_f4(float4* __restrict__ p, int n4) {
    int i = blockIdx.x * 256 + threadIdx.x;
    if (i < n4) p[i] = make_float4(0.f, 0.f, 0.f, 0.f);
}

// ---------------------------------------------------------------------------
// GEMM1: pre1[N,128] = X[N,512] @ W1[512,128]
// grid.x = N/16 row tiles, 256 threads = 8 waves, wave w -> columns [16w,16w+16)
// A tile (16 rows x 64 k) staged in LDS per 64-wide K chunk.
// ---------------------------------------------------------------------------
__global__ __launch_bounds__(256) void gcn_gemm1_wmma(const float* __restrict__ X,
                                                      const float* __restrict__ W1,
                                                      float* __restrict__ pre1,
                                                      int nrows) {
    // stride 68 floats: row base = 272B (16B aligned for float4 staging),
    // row-to-row bank offset = 4 banks -> conflict-free column reads.
    __shared__ float As[16][68];

    const int rowBase = blockIdx.x * 16;
    const int wave    = threadIdx.x >> 5;    // 0..7 : column tile
    const int lane    = threadIdx.x & 31;
    const int half    = lane >> 4;           // 0: K pair {0,1}; 1: K pair {2,3}
    const int l       = lane & 15;
    const int colBase = wave * 16;

    // staging coordinates (256 threads cover 16 rows x 64 floats exactly)
    const int sr = threadIdx.x >> 4;         // 0..15 row
    const int sc = (threadIdx.x & 15) * 4;   // 0..60 col (float4)
    int srow = rowBase + sr; if (srow >= nrows) srow = nrows - 1;   // clamp (N%16==0 anyway)

    v8f c = {};

    for (int kb = 0; kb < NFEAT; kb += 64) {
        // ---- stage A chunk: coalesced float4 global load -> LDS
        const float4 xv = *(const float4*)(X + (size_t)srow * NFEAT + kb + sc);
        *(float4*)(&As[sr][sc]) = xv;
        __syncthreads();

        // ---- 16 WMMA steps of K=4 over this chunk
        #pragma unroll
        for (int kk = 0; kk < 64; kk += 4) {
            // A: 16x4 f32 layout — lane l (half h) supplies row M=l, K = kk+2h, kk+2h+1
            v2f a = *(const v2f*)(&As[l][kk + 2 * half]);
            // B: 4x16 f32 layout — lane l (half h) supplies col N=colBase+l,
            //    rows K = kk+2h and kk+2h+1
            const int krow = kb + kk + 2 * half;
            v2f b;
            b.x = W1[(size_t)(krow + 0) * NHID + colBase + l];
            b.y = W1[(size_t)(krow + 1) * NHID + colBase + l];
            c = __builtin_amdgcn_wmma_f32_16x16x4_f32(false, a, false, b,
                                                      (short)0, c, false, false);
        }
        __syncthreads();
    }

    // ---- store C: lanes 0-15 -> rows v, lanes 16-31 -> rows v+8; col = colBase+l
    #pragma unroll
    for (int v = 0; v < 8; ++v) {
        const int m = v + half * 8;
        if (rowBase + m < nrows)
            pre1[(size_t)(rowBase + m) * NHID + colBase + l] = c[v];
    }
}

// ---------------------------------------------------------------------------
// SpMM layer 1: h[src] += val * pre1[dst]   (128 features, one wave per edge)
// ---------------------------------------------------------------------------
__global__ __launch_bounds__(256) void gcn_spmm1(const int* __restrict__ esrc,
                                                 const int* __restrict__ edst,
                                                 const float* __restrict__ eval,
                                                 const float* __restrict__ pre1,
                                                 float* __restrict__ h,
                                                 int E) {
    const int e = blockIdx.x * 8 + (threadIdx.x >> 5);
    if (e >= E) return;
    const int lane = threadIdx.x & 31;

    const int   s = esrc[e];
    const int   d = edst[e];
    const float v = eval[e];

    const float4 p = ((const float4*)(pre1 + (size_t)d * NHID))[lane];
    float* hr = h + (size_t)s * NHID + lane * 4;
    unsafeAtomicAdd(hr + 0, v * p.x);
    unsafeAtomicAdd(hr + 1, v * p.y);
    unsafeAtomicAdd(hr + 2, v * p.z);
    unsafeAtomicAdd(hr + 3, v * p.w);
}

// ---------------------------------------------------------------------------
// GEMM2: pre2[N,32] = relu(h)[N,128] @ W2[128,32]
// grid.x = N/16 row tiles, 64 threads = 2 waves, wave w -> columns [16w,16w+16)
// ReLU applied on the A-operand load (segment_sum is complete by now).
// ---------------------------------------------------------------------------
__global__ __launch_bounds__(64) void gcn_gemm2_wmma(const float* __restrict__ h,
                                                     const float* __restrict__ W2,
                                                     float* __restrict__ pre2,
                                                     int nrows) {
    const int rowBase = blockIdx.x * 16;
    const int wave    = threadIdx.x >> 5;    // 0..1
    const int lane    = threadIdx.x & 31;
    const int half    = lane >> 4;
    const int l       = lane & 15;
    const int colBase = wave * 16;

    int arow = rowBase + l; if (arow >= nrows) arow = nrows - 1;
    const float* hrow = h + (size_t)arow * NHID;

    v8f c = {};
    #pragma unroll 4
    for (int k = 0; k < NHID; k += 4) {
        v2f a = *(const v2f*)(hrow + k + 2 * half);
        a.x = fmaxf(a.x, 0.f);                       // fused ReLU
        a.y = fmaxf(a.y, 0.f);
        const int krow = k + 2 * half;
        v2f b;
        b.x = W2[(size_t)(krow + 0) * NCLS + colBase + l];
        b.y = W2[(size_t)(krow + 1) * NCLS + colBase + l];
        c = __builtin_amdgcn_wmma_f32_16x16x4_f32(false, a, false, b,
                                                  (short)0, c, false, false);
    }

    #pragma unroll
    for (int v = 0; v < 8; ++v) {
        const int m = v + half * 8;
        if (rowBase + m < nrows)
            pre2[(size_t)(rowBase + m) * NCLS + colBase + l] = c[v];
    }
}

// ---------------------------------------------------------------------------
// SpMM layer 2: out[src] += val * pre2[dst]  (32 features, one wave per edge,
// lane == output class)
// ---------------------------------------------------------------------------
__global__ __launch_bounds__(256) void gcn_spmm2(const int* __restrict__ esrc,
                                                 const int* __restrict__ edst,
                                                 const float* __restrict__ eval,
                                                 const float* __restrict__ pre2,
                                                 float* __restrict__ out,
                                                 int E) {
    const int e = blockIdx.x * 8 + (threadIdx.x >> 5);
    if (e >= E) return;
    const int lane = threadIdx.x & 31;

    const int   s = esrc[e];
    const int   d = edst[e];
    const float v = eval[e];

    const float p = pre2[(size_t)d * NCLS + lane];
    unsafeAtomicAdd(out + (size_t)s * NCLS + lane, v * p);
}

// ---------------------------------------------------------------------------
// launcher
// ---------------------------------------------------------------------------
extern "C" void kernel_launch(void* const* d_in, const int* in_sizes, int n_in,
                              void* d_out, int out_size, void* d_ws, size_t ws_size,
                              hipStream_t stream) {
    const float* X    = (const float*)d_in[0];
    const int*   esrc = (const int*)  d_in[1];
    const int*   edst = (const int*)  d_in[2];
    const float* eval = (const float*)d_in[3];
    const float* W1   = (const float*)d_in[4];
    const float* W2   = (const float*)d_in[5];
    float*       out  = (float*)d_out;

    const int N = in_sizes[0] / NFEAT;   // 50000
    const int E = in_sizes[1];           // 800000

    // workspace: pre1 [N,128] | h [N,128] | pre2 [N,32]  (~57.6 MB)
    float* pre1 = (float*)d_ws;
    float* h    = pre1 + (size_t)N * NHID;
    float* pre2 = h    + (size_t)N * NHID;

    // zero accumulation targets (d_out / d_ws are poisoned by the harness)
    {
        const int n4h = (N * NHID) / 4;
        gcn_zero_f4<<<(n4h + 255) / 256, 256, 0, stream>>>((float4*)h, n4h);
        const int n4o = (N * NCLS) / 4;
        gcn_zero_f4<<<(n4o + 255) / 256, 256, 0, stream>>>((float4*)out, n4o);
    }

    const int rowTiles = (N + 15) / 16;  // 3125

    gcn_gemm1_wmma<<<rowTiles, 256, 0, stream>>>(X, W1, pre1, N);
    gcn_spmm1    <<<(E + 7) / 8, 256, 0, stream>>>(esrc, edst, eval, pre1, h, E);
    gcn_gemm2_wmma<<<rowTiles, 64, 0, stream>>>(h, W2, pre2, N);
    gcn_spmm2    <<<(E + 7) / 8, 256, 0, stream>>>(esrc, edst, eval, pre2, out, E);
}